// PointTriplaneGenerator_15144054686231
// MI455X (gfx1250) — compile-verified
//
#include <hip/hip_runtime.h>
#include <stdint.h>

#define GRID_  128
#define CELLS  (GRID_ * GRID_)     // 16384
#define C_     196
#define N_     131072
#define B_     4
#define CPAD   200                 // padded channel stride for LDS accumulators

// ---------------------------------------------------------------------------
// Zero-fill (float4 vectorized)
// ---------------------------------------------------------------------------
__global__ void zero_f4(float4* __restrict__ p, long n4) {
    long i = (long)blockIdx.x * blockDim.x + threadIdx.x;
    if (i < n4) p[i] = make_float4(0.f, 0.f, 0.f, 0.f);
}

// ---------------------------------------------------------------------------
// Atomic helpers: global (agent scope) -> global_atomic_add_f32,
//                 LDS (workgroup scope on __shared__-derived ptr) -> ds_add_f32
// ---------------------------------------------------------------------------
__device__ __forceinline__ void atom_add(float* p, float v) {
    __hip_atomic_fetch_add(p, v, __ATOMIC_RELAXED, __HIP_MEMORY_SCOPE_AGENT);
}
__device__ __forceinline__ void lds_add(float* p, float v) {
    __hip_atomic_fetch_add(p, v, __ATOMIC_RELAXED, __HIP_MEMORY_SCOPE_WORKGROUP);
}

// ---------------------------------------------------------------------------
// Async prefetch of one point's 196 floats (784B) into LDS, as 2x(32 lanes x
// 16B) = 1024B. The second transfer is per-lane clamped so the tail lanes
// (which only feed unused channels >=196) never read past the input array.
// ---------------------------------------------------------------------------
__device__ __forceinline__ void async_prefetch_point(const float* gp,
                                                     uint32_t lds_base_bytes,
                                                     uint32_t lane,
                                                     const float* gend) {
    uint64_t ga0  = (uint64_t)(uintptr_t)gp + (uint64_t)lane * 16u;
    uint64_t ga1  = ga0 + 512u;
    uint64_t gmax = (uint64_t)(uintptr_t)gend - 16u;
    if (ga1 > gmax) ga1 = gmax;          // only affects lanes feeding c >= 196
    uint32_t l0 = lds_base_bytes + lane * 16u;
    uint32_t l1 = l0 + 512u;
    asm volatile("global_load_async_to_lds_b128 %0, %1, off"
                 :: "v"(l0), "v"(ga0) : "memory");
    asm volatile("global_load_async_to_lds_b128 %0, %1, off"
                 :: "v"(l1), "v"(ga1) : "memory");
}

// ---------------------------------------------------------------------------
// Scatter kernel: grid = (512 blocks, B_ batches), 256 threads = 8 waves.
// Double-buffered async global->LDS staging of point features; the 4 corner
// cells of each plane (structural hot spots created by clipping) are
// privatized per-workgroup in LDS via ds_add_f32 and flushed once at the end.
// ---------------------------------------------------------------------------
__global__ void __launch_bounds__(256)
splat_kernel(const float* __restrict__ feats, const float* __restrict__ sb,
             float* __restrict__ out, float* __restrict__ den) {
    __shared__ float staging[8][2][256];        // 8 waves x 2 bufs x 1KB
    __shared__ float acc[3][4][CPAD];           // plane x corner x channel
    __shared__ float accDen[3][4];              // plane x corner

    const int lane = threadIdx.x & 31;
    const int wv   = threadIdx.x >> 5;
    const int b    = blockIdx.y;
    const int wgid = blockIdx.x * 8 + wv;       // global wave id in batch
    const int strd = gridDim.x * 8;

    // zero LDS corner accumulators
    for (int idx = threadIdx.x; idx < 3 * 4 * CPAD; idx += 256)
        acc[0][0][idx] = 0.f;
    if (threadIdx.x < 12) accDen[0][threadIdx.x] = 0.f;

    // scene bounds -> normalization scale
    const float lo0 = sb[0], hi0 = sb[1];
    const float lo1 = sb[2], hi1 = sb[3];
    const float lo2 = sb[4], hi2 = sb[5];
    const float s0 = 2.0f / (hi0 - lo0);
    const float s1 = 2.0f / (hi1 - lo1);
    const float s2 = 2.0f / (hi2 - lo2);

    const float* batchBase = feats + (size_t)b * N_ * C_;
    const float* gend      = feats + (size_t)B_ * N_ * C_;

    // Low 32 bits of the LDS-aperture flat address == raw LDS byte offset.
    const uint32_t ldsWave = (uint32_t)(uintptr_t)(&staging[wv][0][0]);

    const bool active = (wgid < N_);
    int buf = 0;
    if (active)
        async_prefetch_point(batchBase + (size_t)wgid * C_, ldsWave, lane, gend);

    __syncthreads();   // acc[] zeroed before any wave scatters into it

    const size_t ob = (size_t)b * 3 * C_ * CELLS;

    for (int i = wgid; i < N_; i += strd) {
        const int inext = i + strd;
        if (inext < N_) {
            async_prefetch_point(batchBase + (size_t)inext * C_,
                                 ldsWave + (uint32_t)((buf ^ 1) * 1024), lane, gend);
            asm volatile("s_wait_asynccnt 2" ::: "memory");   // current buf done
        } else {
            asm volatile("s_wait_asynccnt 0" ::: "memory");
        }

        const float* sbuf = &staging[wv][buf][0];

        // header broadcast: x, y, z, f3 (same LDS address for all lanes)
        float4 h = *(const float4*)sbuf;
        const float cn0 = (h.x - lo0) * s0 - 1.0f;
        const float cn1 = (h.y - lo1) * s1 - 1.0f;
        const float cn2 = (h.z - lo2) * s2 - 1.0f;
        const float alpha = 1.0f / (1.0f + __expf(-h.w));

        int g0 = (int)((cn0 * 0.5f + 0.5f) * (float)(GRID_ - 1));
        int g1 = (int)((cn1 * 0.5f + 0.5f) * (float)(GRID_ - 1));
        int g2 = (int)((cn2 * 0.5f + 0.5f) * (float)(GRID_ - 1));
        g0 = min(max(g0, 0), GRID_ - 1);
        g1 = min(max(g1, 0), GRID_ - 1);
        g2 = min(max(g2, 0), GRID_ - 1);
        const int cell0 = g0 * GRID_ + g1;        // xy plane
        const int cell1 = g0 * GRID_ + g2;        // xz plane
        const int cell2 = g1 * GRID_ + g2;        // yz plane

        // corner detection (wave-uniform): both axes clipped to an extreme
        const bool e00 = (g0 == 0) | (g0 == GRID_ - 1);
        const bool e11 = (g1 == 0) | (g1 == GRID_ - 1);
        const bool e22 = (g2 == 0) | (g2 == GRID_ - 1);
        const bool crn0 = e00 & e11, crn1 = e00 & e22, crn2 = e11 & e22;
        const int  ci0 = ((g0 == GRID_ - 1) ? 2 : 0) + ((g1 == GRID_ - 1) ? 1 : 0);
        const int  ci1 = ((g0 == GRID_ - 1) ? 2 : 0) + ((g2 == GRID_ - 1) ? 1 : 0);
        const int  ci2 = ((g1 == GRID_ - 1) ? 2 : 0) + ((g2 == GRID_ - 1) ? 1 : 0);

        if (lane == 0) {
            const int dbase = (b * 3) << 14;
            if (crn0) lds_add(&accDen[0][ci0], alpha);
            else      atom_add(den + dbase + cell0, alpha);
            if (crn1) lds_add(&accDen[1][ci1], alpha);
            else      atom_add(den + dbase + (1 << 14) + cell1, alpha);
            if (crn2) lds_add(&accDen[2][ci2], alpha);
            else      atom_add(den + dbase + (2 << 14) + cell2, alpha);
        }

        float* o0 = out + ob + cell0;
        float* o1 = out + ob + (size_t)C_ * CELLS + cell1;
        float* o2 = out + ob + (size_t)2 * C_ * CELLS + cell2;

        // pass 0: channels lane*4 .. lane*4+3  (0..127)
        {
            float4 v = *(const float4*)(sbuf + lane * 4);
            if (lane == 0) { v.x = cn0; v.y = cn1; v.z = cn2; }  // feats[0..2]=coords_n
            const float vv[4] = {v.x, v.y, v.z, v.w};
            const int c0 = lane * 4;
            #pragma unroll
            for (int j = 0; j < 4; ++j) {
                const float val = vv[j] * alpha;
                const int   c   = c0 + j;
                if (crn0) lds_add(&acc[0][ci0][c], val);
                else      atom_add(o0 + (size_t)c * CELLS, val);
                if (crn1) lds_add(&acc[1][ci1][c], val);
                else      atom_add(o1 + (size_t)c * CELLS, val);
                if (crn2) lds_add(&acc[2][ci2][c], val);
                else      atom_add(o2 + (size_t)c * CELLS, val);
            }
        }
        // pass 1: channels 128 + lane*4  (128..195; lanes >= 17 idle)
        {
            const int c0 = 128 + lane * 4;
            if (c0 < C_) {
                float4 v = *(const float4*)(sbuf + 128 + lane * 4);
                const float vv[4] = {v.x, v.y, v.z, v.w};
                #pragma unroll
                for (int j = 0; j < 4; ++j) {
                    const float val = vv[j] * alpha;
                    const int   c   = c0 + j;
                    if (crn0) lds_add(&acc[0][ci0][c], val);
                    else      atom_add(o0 + (size_t)c * CELLS, val);
                    if (crn1) lds_add(&acc[1][ci1][c], val);
                    else      atom_add(o1 + (size_t)c * CELLS, val);
                    if (crn2) lds_add(&acc[2][ci2][c], val);
                    else      atom_add(o2 + (size_t)c * CELLS, val);
                }
            }
        }
        buf ^= 1;
    }

    // ------- flush privatized corner accumulators (one atomic per WG each) --
    __syncthreads();
    for (int idx = threadIdx.x; idx < 3 * 4 * C_; idx += 256) {
        const int p      = idx / (4 * C_);
        const int r      = idx - p * (4 * C_);
        const int corner = r / C_;
        const int c      = r - corner * C_;
        const float v    = acc[p][corner][c];
        if (v != 0.f) {
            const int gA   = (corner >> 1) * (GRID_ - 1);
            const int gB   = (corner & 1)  * (GRID_ - 1);
            const int cell = gA * GRID_ + gB;
            atom_add(out + ob + ((size_t)p * C_ + c) * CELLS + cell, v);
        }
    }
    if (threadIdx.x < 12) {
        const int p      = threadIdx.x >> 2;
        const int corner = threadIdx.x & 3;
        const float v    = accDen[p][corner];
        if (v != 0.f) {
            const int gA   = (corner >> 1) * (GRID_ - 1);
            const int gB   = (corner & 1)  * (GRID_ - 1);
            const int cell = gA * GRID_ + gB;
            atom_add(den + ((b * 3 + p) << 14) + cell, v);
        }
    }
}

// ---------------------------------------------------------------------------
// Normalize: out[b,p,c,cell] *= 1/max(den[b,p,cell], 1e-6)
// ---------------------------------------------------------------------------
__global__ void normalize_kernel(float* __restrict__ out,
                                 const float* __restrict__ den) {
    const int t = blockIdx.x * blockDim.x + threadIdx.x;
    if (t >= B_ * 3 * CELLS) return;
    const int   bp  = t >> 14;
    const float inv = 1.0f / fmaxf(den[t], 1e-6f);
    float* base = out + (size_t)bp * C_ * CELLS + (t & (CELLS - 1));
    #pragma unroll 4
    for (int c = 0; c < C_; ++c)
        base[(size_t)c * CELLS] *= inv;
}

// ---------------------------------------------------------------------------
extern "C" void kernel_launch(void* const* d_in, const int* in_sizes, int n_in,
                              void* d_out, int out_size, void* d_ws, size_t ws_size,
                              hipStream_t stream) {
    (void)in_sizes; (void)n_in; (void)ws_size;
    const float* feats = (const float*)d_in[0];   // [4,131072,196] f32
    const float* sb    = (const float*)d_in[1];   // [6] f32
    float* out = (float*)d_out;                   // [4,3,196,128,128] f32
    float* den = (float*)d_ws;                    // [4,3,16384] f32

    // zero numerators (accumulated in-place in d_out) and denominators
    const long n4o = (long)out_size / 4;
    zero_f4<<<(unsigned)((n4o + 255) / 256), 256, 0, stream>>>((float4*)out, n4o);
    const long n4d = (long)(B_ * 3 * CELLS) / 4;
    zero_f4<<<(unsigned)((n4d + 255) / 256), 256, 0, stream>>>((float4*)den, n4d);

    // scatter: 512 blocks x 8 waves per batch -> 32 points per wave
    splat_kernel<<<dim3(512, B_), 256, 0, stream>>>(feats, sb, out, den);

    // normalize
    const int nt = B_ * 3 * CELLS;
    normalize_kernel<<<(nt + 255) / 256, 256, 0, stream>>>(out, den);
}